// MetaPathLearner_35304631173350
// MI455X (gfx1250) — compile-verified
//
#include <hip/hip_runtime.h>
#include <hip/hip_bf16.h>
#include <stdint.h>

#define NUM_CHANNELS   4
#define NUM_EDGE_TYPES 8
#define EDGES_PER_TYPE (1 << 18)          // 262144
#define BLOCKS_PER_TYPE (EDGES_PER_TYPE / 256)   // 1024
#define NUM_NODES      4096
#define NN             ((size_t)NUM_NODES * (size_t)NUM_NODES)     // 16,777,216
#define TOTAL_EDGES    ((long long)NUM_EDGE_TYPES * EDGES_PER_TYPE) // 2,097,152

// ---------------------------------------------------------------------------
// Kernel 1: softmax over edge-type axis of the 4x8 weight matrix.
// Writes soft weights both to the output tail and to workspace (read by scatter).
// ---------------------------------------------------------------------------
__global__ void mpl_softmax_kernel(const float* __restrict__ w,
                                   float* __restrict__ sw_ws,
                                   float* __restrict__ sw_out) {
    int c = threadIdx.x;
    if (c < NUM_CHANNELS) {
        float row[NUM_EDGE_TYPES];
        float m = -INFINITY;
        #pragma unroll
        for (int e = 0; e < NUM_EDGE_TYPES; ++e) {
            row[e] = w[c * NUM_EDGE_TYPES + e];
            m = fmaxf(m, row[e]);
        }
        float s = 0.0f;
        #pragma unroll
        for (int e = 0; e < NUM_EDGE_TYPES; ++e) {
            row[e] = __expf(row[e] - m);
            s += row[e];
        }
        float inv = 1.0f / s;
        #pragma unroll
        for (int e = 0; e < NUM_EDGE_TYPES; ++e) {
            float v = row[e] * inv;
            sw_ws[c * NUM_EDGE_TYPES + e]  = v;
            sw_out[c * NUM_EDGE_TYPES + e] = v;
        }
    }
}

// ---------------------------------------------------------------------------
// Kernel 2: zero-fill one 64 MB channel slab with 128-bit stores.
// Leaves the slab hot in the 192 MB L2 so the following atomics hit L2.
// ---------------------------------------------------------------------------
__global__ void mpl_zero_kernel(float4* __restrict__ out) {
    size_t i = (size_t)blockIdx.x * blockDim.x + threadIdx.x;
    out[i] = make_float4(0.f, 0.f, 0.f, 0.f);
}

// ---------------------------------------------------------------------------
// Kernel 3: fused scatter for one channel.
// Each block handles 256 consecutive edges of a single edge type; edge type is
// a pure function of blockIdx.x so the softmax weight fetch is scalar.
// Edge indices are staged into LDS with CDNA5 async global->LDS copies
// (ASYNCcnt-tracked), then each lane issues one native, non-returning
// global_atomic_add_f32 at device scope into the channel slab.
// ---------------------------------------------------------------------------
__global__ void mpl_scatter_kernel(const long long* __restrict__ ei,
                                   const float* __restrict__ sw,
                                   float* __restrict__ out,
                                   int channel) {
    __shared__ long long s_src[256];
    __shared__ long long s_dst[256];

    const int tid = threadIdx.x;
    const int bid = blockIdx.x;                      // 0 .. 8191
    const int e   = bid >> 10;                       // uniform edge type (1024 blocks/type)
    const int j   = ((bid & (BLOCKS_PER_TYPE - 1)) << 8) + tid;  // edge within type

    // flat index into [8, 2, 262144] int64 (fits in 32 bits)
    const int idx_src = (e * 2) * EDGES_PER_TYPE + j;
    const long long* srcp = ei + idx_src;
    const long long* dstp = srcp + EDGES_PER_TYPE;

    // LDS byte offsets for this lane (low 32 bits of a generic LDS pointer are
    // the workgroup-relative LDS address used by DS/async instructions).
    unsigned lds_src = (unsigned)(uintptr_t)(void*)&s_src[tid];
    unsigned lds_dst = (unsigned)(uintptr_t)(void*)&s_dst[tid];

    // CDNA5 async copies: per-lane 8B global -> LDS, tracked with ASYNCcnt.
    asm volatile(
        "global_load_async_to_lds_b64 %0, %1, off\n\t"
        "global_load_async_to_lds_b64 %2, %3, off"
        :
        : "v"(lds_src), "v"(srcp), "v"(lds_dst), "v"(dstp)
        : "memory");
    asm volatile("s_wait_asynccnt 0" ::: "memory");

    const int src = (int)s_src[tid];
    const int dst = (int)s_dst[tid];

    const float wv = sw[channel * NUM_EDGE_TYPES + e];   // uniform -> scalar load
    float* addr = out + (size_t)((src << 12) + dst);     // slab base passed in; N = 4096

    // Native non-returning f32 atomic add, device scope (RMW at L2).
    asm volatile("global_atomic_add_f32 %0, %1, off scope:SCOPE_DEV"
                 :
                 : "v"(addr), "v"(wv)
                 : "memory");
}

// ---------------------------------------------------------------------------
// Launch: softmax once, then per channel: zero slab -> scatter into slab.
// Channel-serial ordering keeps each 64 MB slab resident in L2 across
// zero-fill and the 2.1M atomics that follow.
// ---------------------------------------------------------------------------
extern "C" void kernel_launch(void* const* d_in, const int* in_sizes, int n_in,
                              void* d_out, int out_size, void* d_ws, size_t ws_size,
                              hipStream_t stream) {
    const float*      weights = (const float*)d_in[0];
    const long long*  ei      = (const long long*)d_in[1];   // int64 [8, 2, 262144]
    float*            out     = (float*)d_out;               // [4*N*N] ++ [32]
    float*            sw_ws   = (float*)d_ws;                // 32 floats scratch
    float*            sw_out  = out + (size_t)NUM_CHANNELS * NN;

    mpl_softmax_kernel<<<1, 32, 0, stream>>>(weights, sw_ws, sw_out);

    const int zero_blocks    = (int)(NN / 4 / 256);            // 16384
    const int scatter_blocks = (int)(TOTAL_EDGES / 256);       // 8192

    for (int c = 0; c < NUM_CHANNELS; ++c) {
        float* slab = out + (size_t)c * NN;
        mpl_zero_kernel<<<zero_blocks, 256, 0, stream>>>(
            reinterpret_cast<float4*>(slab));
        mpl_scatter_kernel<<<scatter_blocks, 256, 0, stream>>>(
            ei, sw_ws, slab, c);
    }
}